// DIN_49383533969756
// MI455X (gfx1250) — compile-verified
//
#include <hip/hip_runtime.h>
#include <math.h>

// ---------------------------------------------------------------------------
// DIN recommender for MI455X (gfx1250, wave32).
// Attention MLP is linear -> collapsed to a single 384-vector (exact algebra).
// Dense layers use V_WMMA_F32_16X16X4_F32 (full f32 precision on matrix core).
// ---------------------------------------------------------------------------

typedef __attribute__((ext_vector_type(2))) float v2f;
typedef __attribute__((ext_vector_type(8))) float v8f;

#define BATCH   2048
#define LHIST   200
#define EPS_D   1e-8f

__device__ __forceinline__ float waveReduceSum(float v) {
#pragma unroll
  for (int off = 16; off > 0; off >>= 1)
    v += __shfl_xor(v, off, 32);
  return v;
}

// --- Kernel 1: collapse attention MLP: v[0:384] = w_att2 @ w_att1,
//     v[384] = w_att2 @ b_att1 + b_att2 ------------------------------------
__global__ void precompute_v(const float* __restrict__ w_att1,
                             const float* __restrict__ b_att1,
                             const float* __restrict__ w_att2,
                             const float* __restrict__ b_att2,
                             float* __restrict__ v) {
  int tid = threadIdx.x;
  if (tid < 384) {
    float acc = 0.f;
#pragma unroll
    for (int i = 0; i < 36; ++i)
      acc += w_att2[i] * w_att1[i * 384 + tid];
    v[tid] = acc;
  } else if (tid == 384) {
    float acc = b_att2[0];
#pragma unroll
    for (int i = 0; i < 36; ++i)
      acc += w_att2[i] * b_att1[i];
    v[384] = acc;
  }
}

// --- Kernel 2: one wave per batch row: gather embeddings, attention scores,
//     browse accumulation, assemble feat = [item_emb(96), browse(96), user(64)]
__global__ void attention_feat(const int* __restrict__ user,
                               const int* __restrict__ item,
                               const int* __restrict__ rec_his,
                               const float* __restrict__ t_user,
                               const float* __restrict__ t1,
                               const float* __restrict__ t2,
                               const float* __restrict__ t3,
                               const float* __restrict__ v,
                               float* __restrict__ feat) {
  int b    = blockIdx.x * (blockDim.x >> 5) + (threadIdx.x >> 5);
  int lane = threadIdx.x & 31;
  if (b >= BATCH) return;

  const int* itm = item + b * 3;
  int i0 = itm[0], i1 = itm[1], i2 = itm[2];

  // lane owns features f = lane, lane+32, lane+64 of the 96-dim vectors
  float q0 = t1[i0 * 32 + lane];
  float q1 = t2[i1 * 32 + lane];
  float q2 = t3[i2 * 32 + lane];

  float vq0 = v[lane],       vq1 = v[32 + lane],  vq2 = v[64 + lane];
  float vu0 = v[96 + lane],  vu1 = v[128 + lane], vu2 = v[160 + lane];
  float vd0 = v[192 + lane], vd1 = v[224 + lane], vd2 = v[256 + lane];
  float vp0 = v[288 + lane], vp1 = v[320 + lane], vp2 = v[352 + lane];
  float c   = v[384];

  float wb0 = (vu0 - vd0) + vp0 * q0;
  float wb1 = (vu1 - vd1) + vp1 * q1;
  float wb2 = (vu2 - vd2) + vp2 * q2;
  float s0  = waveReduceSum((vq0 + vd0) * q0 + (vq1 + vd1) * q1 +
                            (vq2 + vd2) * q2) + c;

  float br0 = 0.f, br1 = 0.f, br2 = 0.f;
  const int* rh = rec_his + (size_t)b * LHIST * 3;
  for (int l = 0; l < LHIST; ++l) {
    int j0 = rh[l * 3 + 0];
    int j1 = rh[l * 3 + 1];
    int j2 = rh[l * 3 + 2];
    float u0 = t1[j0 * 32 + lane];
    float u1 = t2[j1 * 32 + lane];
    float u2 = t3[j2 * 32 + lane];
    float s  = s0 + waveReduceSum(wb0 * u0 + wb1 * u1 + wb2 * u2);
    if (j0 == 0) s = 0.f;                 // mask: rec_his[:,:,0]==0
    br0 += s * u0; br1 += s * u1; br2 += s * u2;
  }

  float* fb = feat + (size_t)b * 256;
  fb[lane]       = q0;  fb[32 + lane]  = q1;  fb[64 + lane]  = q2;
  fb[96 + lane]  = br0; fb[128 + lane] = br1; fb[160 + lane] = br2;
  int uidx = user[b];
  fb[192 + lane] = t_user[uidx * 64 + lane];
  fb[224 + lane] = t_user[uidx * 64 + 32 + lane];
}

// --- Kernel 3/6: f32 WMMA GEMM: C(MxN) = A(MxK) @ W(N,K)^T + bias ----------
// One wave per 16x16 C tile, K-loop of V_WMMA_F32_16X16X4_F32.
__global__ void gemm_wmma_f32(const float* __restrict__ A, int lda,
                              const float* __restrict__ W, int K,
                              const float* __restrict__ bias,
                              float* __restrict__ C, int N,
                              int mTiles, int nTiles) {
  int wave  = (blockIdx.x * blockDim.x + threadIdx.x) >> 5;
  int lane  = threadIdx.x & 31;
  int mTile = wave / nTiles;
  int nTile = wave - mTile * nTiles;
  if (mTile >= mTiles) return;        // wave-uniform: EXEC all-1s inside

  int m0 = mTile << 4, n0 = nTile << 4;
  int laneLo = lane & 15;
  int hiK    = (lane >> 4) << 1;      // lanes 16-31 carry K=2,3 of each step

  // A 16x4 frag: lane -> row m0+laneLo, cols k+hiK, k+hiK+1 (contig float2)
  const float* aPtr = A + (size_t)(m0 + laneLo) * lda + hiK;
  // B 4x16 frag: B[kk][n] = W[n*K+kk]; lane -> col n0+laneLo, rows k+hiK(+1)
  int  nCol   = n0 + laneLo;
  bool nValid = nCol < N;
  const float* bPtr = W + (size_t)(nValid ? nCol : 0) * K + hiK;

  v8f acc = {};
  for (int k = 0; k < K; k += 4) {
    v2f a = *(const v2f*)(aPtr + k);
    v2f b = *(const v2f*)(bPtr + k);
    acc = __builtin_amdgcn_wmma_f32_16x16x4_f32(
        /*neg_a=*/false, a, /*neg_b=*/false, b,
        /*c_mod=*/(short)0, acc, /*reuse_a=*/false, /*reuse_b=*/false);
  }

  if (nValid) {
    float bn = bias[nCol];
    int mBase = m0 + ((lane >> 4) << 3);  // hi half-wave holds rows M+8
#pragma unroll
    for (int r = 0; r < 8; ++r)
      C[(size_t)(mBase + r) * N + nCol] = acc[r] + bn;
  }
}

// --- Kernel 4/7: per-column mean + rstd (ddof=1) over the batch ------------
__global__ void colstats(const float* __restrict__ x, int rows, int cols,
                         float* __restrict__ meanOut,
                         float* __restrict__ rstdOut) {
  int col = blockIdx.x;
  __shared__ float ssum[256], ssq[256];
  float sum = 0.f, sq = 0.f;
  for (int r = threadIdx.x; r < rows; r += blockDim.x) {
    float v = x[(size_t)r * cols + col];
    sum += v; sq += v * v;
  }
  ssum[threadIdx.x] = sum; ssq[threadIdx.x] = sq;
  __syncthreads();
  for (int s = 128; s > 0; s >>= 1) {
    if (threadIdx.x < s) {
      ssum[threadIdx.x] += ssum[threadIdx.x + s];
      ssq[threadIdx.x]  += ssq[threadIdx.x + s];
    }
    __syncthreads();
  }
  if (threadIdx.x == 0) {
    float mean = ssum[0] / (float)rows;
    float var  = (ssq[0] - ssum[0] * mean) / (float)(rows - 1);
    meanOut[col] = mean;
    rstdOut[col] = rsqrtf(var + EPS_D);
  }
}

// --- Kernel 5: DICE activation, in-place -----------------------------------
__global__ void dice_apply(float* __restrict__ x, int n, int cols,
                           const float* __restrict__ mean,
                           const float* __restrict__ rstd,
                           const float* __restrict__ alpha,
                           const float* __restrict__ gamma,
                           const float* __restrict__ beta) {
  int i = blockIdx.x * blockDim.x + threadIdx.x;
  if (i >= n) return;
  int col  = i % cols;
  float xv = x[i];
  float xn = (xv - mean[col]) * rstd[col] * gamma[col] + beta[col];
  float p  = 1.f / (1.f + expf(-xn));
  x[i] = xn * (p + alpha[col] * (1.f - p));
}

// --- Kernel 8: DICE(layer2) fused with 80->1 matvec + sigmoid --------------
__global__ void final_head(const float* __restrict__ z2,
                           const float* __restrict__ mean,
                           const float* __restrict__ rstd,
                           const float* __restrict__ a2,
                           const float* __restrict__ g2,
                           const float* __restrict__ be2,
                           const float* __restrict__ w3,
                           const float* __restrict__ b3,
                           float* __restrict__ out) {
  int b    = blockIdx.x * (blockDim.x >> 5) + (threadIdx.x >> 5);
  int lane = threadIdx.x & 31;
  if (b >= BATCH) return;
  const float* row = z2 + (size_t)b * 80;
  float acc = 0.f;
  for (int n = lane; n < 80; n += 32) {
    float xv = row[n];
    float xn = (xv - mean[n]) * rstd[n] * g2[n] + be2[n];
    float p  = 1.f / (1.f + expf(-xn));
    acc += xn * (p + a2[n] * (1.f - p)) * w3[n];
  }
  acc = waveReduceSum(acc);
  if (lane == 0) out[b] = 1.f / (1.f + expf(-(acc + b3[0])));
}

// ---------------------------------------------------------------------------
extern "C" void kernel_launch(void* const* d_in, const int* in_sizes, int n_in,
                              void* d_out, int out_size, void* d_ws, size_t ws_size,
                              hipStream_t stream) {
  const int*   user    = (const int*)d_in[0];
  const int*   item    = (const int*)d_in[1];
  const int*   rec_his = (const int*)d_in[2];
  const float* t_user  = (const float*)d_in[3];
  const float* t1      = (const float*)d_in[4];
  const float* t2      = (const float*)d_in[5];
  const float* t3      = (const float*)d_in[6];
  const float* w_att1  = (const float*)d_in[7];
  const float* b_att1  = (const float*)d_in[8];
  const float* w_att2  = (const float*)d_in[9];
  const float* b_att2  = (const float*)d_in[10];
  const float* w1      = (const float*)d_in[11];
  const float* b1      = (const float*)d_in[12];
  const float* a1      = (const float*)d_in[13];
  const float* g1      = (const float*)d_in[14];
  const float* be1     = (const float*)d_in[15];
  const float* w2      = (const float*)d_in[16];
  const float* b2      = (const float*)d_in[17];
  const float* a2      = (const float*)d_in[18];
  const float* g2      = (const float*)d_in[19];
  const float* be2     = (const float*)d_in[20];
  const float* w3      = (const float*)d_in[21];
  const float* b3      = (const float*)d_in[22];
  float* out = (float*)d_out;

  // workspace layout (floats)
  float* ws    = (float*)d_ws;
  float* v     = ws;                    // 385   (pad 512)
  float* feat  = ws + 512;              // 2048*256 = 524288
  float* z1    = feat + 524288;         // 2048*200 = 409600
  float* s1m   = z1 + 409600;           // 200
  float* s1r   = s1m + 256;             // 200   (pad)
  float* z2    = s1r + 256;             // 2048*80 = 163840
  float* s2m   = z2 + 163840;           // 80
  float* s2r   = s2m + 128;             // 80

  // 1) collapse the (linear) attention MLP
  precompute_v<<<1, 512, 0, stream>>>(w_att1, b_att1, w_att2, b_att2, v);

  // 2) gathers + attention + browse + feat assembly (1 wave per row)
  attention_feat<<<BATCH / 8, 256, 0, stream>>>(user, item, rec_his, t_user,
                                                t1, t2, t3, v, feat);

  // 3) z1 = feat(2048x256) @ w1^T + b1   (M=2048,N=200,K=256)
  {
    int mT = BATCH / 16, nT = (200 + 15) / 16;        // 128 x 13 tiles
    int blocks = (mT * nT + 7) / 8;                   // 8 waves / block
    gemm_wmma_f32<<<blocks, 256, 0, stream>>>(feat, 256, w1, 256, b1,
                                              z1, 200, mT, nT);
  }
  // 4) batch stats + 5) DICE in-place
  colstats<<<200, 256, 0, stream>>>(z1, BATCH, 200, s1m, s1r);
  dice_apply<<<(BATCH * 200 + 255) / 256, 256, 0, stream>>>(
      z1, BATCH * 200, 200, s1m, s1r, a1, g1, be1);

  // 6) z2 = z1(2048x200) @ w2^T + b2   (M=2048,N=80,K=200)
  {
    int mT = BATCH / 16, nT = 80 / 16;                // 128 x 5 tiles
    int blocks = (mT * nT + 7) / 8;
    gemm_wmma_f32<<<blocks, 256, 0, stream>>>(z1, 200, w2, 200, b2,
                                              z2, 80, mT, nT);
  }
  // 7) batch stats + 8) DICE + head matvec + sigmoid
  colstats<<<80, 256, 0, stream>>>(z2, BATCH, 80, s2m, s2r);
  final_head<<<BATCH / 8, 256, 0, stream>>>(z2, s2m, s2r, a2, g2, be2,
                                            w3, b3, out);
}